// O2OClsHead_36318243455494
// MI455X (gfx1250) — compile-verified
//
#include <hip/hip_runtime.h>
#include <math.h>
#include <stdint.h>

typedef float v2f __attribute__((ext_vector_type(2)));
typedef float v8f __attribute__((ext_vector_type(8)));
typedef int   v4i __attribute__((ext_vector_type(4)));

typedef __attribute__((address_space(1))) v4i* gptr_v4i;
typedef __attribute__((address_space(3))) v4i* lptr_v4i;

#define B_SZ 4
#define N_SZ 512
#define HD   64
#define ID   128
#define PI_F 3.14159265358979323846f

#if __has_builtin(__builtin_amdgcn_global_load_async_to_lds_b128)
#define HAVE_ASYNC_LDS 1
#else
#define HAVE_ASYNC_LDS 0
#endif

__device__ __forceinline__ void wait_async0() {
#if __has_builtin(__builtin_amdgcn_s_wait_asynccnt)
  __builtin_amdgcn_s_wait_asynccnt(0);
#else
  asm volatile("s_wait_asynccnt 0x0" ::: "memory");
#endif
}

__device__ __forceinline__ v8f wmma_f32(v2f a, v2f b, v8f c) {
  // V_WMMA_F32_16X16X4_F32 : D = A(16x4 f32) x B(4x16 f32) + C(16x16 f32)
  return __builtin_amdgcn_wmma_f32_16x16x4_f32(false, a, false, b, (short)0, c, false, false);
}

__device__ __forceinline__ float gelu_exact(float x) {
  return 0.5f * x * (1.0f + erff(x * 0.70710678118654752f));
}

__global__ void init_kernel(int* __restrict__ nm) {
  int idx = blockIdx.x * blockDim.x + threadIdx.x;
  if (idx < B_SZ * N_SZ) nm[idx] = 0;  // 0.0f bits
}

// ---------------------------------------------------------------------------
// prep: feats = relu(X@W_cls+b_cls); u/v = feats@W_{in,out} + pos@W_pos + bias;
//       A = u@W_e1 + b_e1 ; C = v@W_e1.   One wave handles 16 rows end-to-end.
// ---------------------------------------------------------------------------
__global__ __launch_bounds__(128)
void prep_kernel(const float* __restrict__ X, const float* __restrict__ emb,
                 const float* __restrict__ W_cls, const float* __restrict__ b_cls,
                 const float* __restrict__ W_pos, const float* __restrict__ b_pos,
                 const float* __restrict__ W_in,  const float* __restrict__ b_in,
                 const float* __restrict__ W_out, const float* __restrict__ b_out,
                 const float* __restrict__ W_e1,  const float* __restrict__ b_e1,
                 float* __restrict__ Amat, float* __restrict__ Cmat)
{
  __shared__ float Fs[4][16 * HD];   // per-wave feats tile  (16x64)
  __shared__ float Us[4][16 * ID];   // per-wave u/v tile    (16x128)

  const int wave = threadIdx.x >> 5;
  const int lane = threadIdx.x & 31;
  const int row  = lane & 15;   // M index for A-frags, N index for B/D frags
  const int hi   = lane >> 4;
  const int kb   = hi * 2;      // K base within 4-wide K step
  const int tile = blockIdx.x * 4 + wave;
  const int b    = tile >> 5;            // 32 row-tiles per batch
  const int i0   = (tile & 31) << 4;

  float* F = Fs[wave];
  float* U = Us[wave];

  // ---- feats tile: relu(X @ W_cls + b_cls) --------------------------------
  for (int nt = 0; nt < 4; ++nt) {
    v8f acc = {0.f, 0.f, 0.f, 0.f, 0.f, 0.f, 0.f, 0.f};
#pragma unroll
    for (int kk = 0; kk < HD; kk += 4) {
      v2f a; a.x = X[(b * N_SZ + i0 + row) * HD + kk + kb];
             a.y = X[(b * N_SZ + i0 + row) * HD + kk + kb + 1];
      v2f w; w.x = W_cls[(kk + kb    ) * HD + nt * 16 + row];
             w.y = W_cls[(kk + kb + 1) * HD + nt * 16 + row];
      acc = wmma_f32(a, w, acc);
    }
#pragma unroll
    for (int r = 0; r < 8; ++r) {
      int m = r + hi * 8, n = nt * 16 + row;
      F[m * HD + n] = fmaxf(acc[r] + b_cls[n], 0.f);
    }
  }

  // ---- per-row sample-point positions (the only use of tan/cos) -----------
  float p0[8], p1[8];
  const float lin1 = (1.0f - 1e-5f) / 71.0f;
  const float y1v  = 160.f - lin1 * 320.f;
#pragma unroll
  for (int r = 0; r < 8; ++r) {
    int gi = i0 + r + hi * 8;
    float e0 = emb[(b * N_SZ + gi) * 2 + 0];
    float e1 = emb[(b * N_SZ + gi) * 2 + 1];
    float ang = e0 * PI_F;
    float t  = tanf(ang);
    float rc = (e1 * 800.f) / cosf(ang);
    p0[r] = (-t * 160.f + rc) * (1.f / 800.f);
    p1[r] = (-t * y1v  + rc) * (1.f / 800.f);
  }

  // ---- u = feats@W_in + pos@W_pos + (b_in + b_pos) -> U -------------------
  for (int nt = 0; nt < 8; ++nt) {
    v8f acc = {0.f, 0.f, 0.f, 0.f, 0.f, 0.f, 0.f, 0.f};
#pragma unroll
    for (int kk = 0; kk < HD; kk += 4) {
      v2f a; a.x = F[row * HD + kk + kb]; a.y = F[row * HD + kk + kb + 1];
      v2f w; w.x = W_in[(kk + kb    ) * ID + nt * 16 + row];
             w.y = W_in[(kk + kb + 1) * ID + nt * 16 + row];
      acc = wmma_f32(a, w, acc);
    }
#pragma unroll
    for (int r = 0; r < 8; ++r) {
      int m = r + hi * 8, n = nt * 16 + row;
      U[m * ID + n] = acc[r] + b_in[n] + b_pos[n]
                    + p0[r] * W_pos[n] + p1[r] * W_pos[ID + n];
    }
  }

  // ---- A = U @ W_e1 + b_e1 -----------------------------------------------
  for (int nt = 0; nt < 8; ++nt) {
    v8f acc = {0.f, 0.f, 0.f, 0.f, 0.f, 0.f, 0.f, 0.f};
#pragma unroll
    for (int kk = 0; kk < ID; kk += 4) {
      v2f a; a.x = U[row * ID + kk + kb]; a.y = U[row * ID + kk + kb + 1];
      v2f w; w.x = W_e1[(kk + kb    ) * ID + nt * 16 + row];
             w.y = W_e1[(kk + kb + 1) * ID + nt * 16 + row];
      acc = wmma_f32(a, w, acc);
    }
#pragma unroll
    for (int r = 0; r < 8; ++r) {
      int m = r + hi * 8, n = nt * 16 + row;
      Amat[(b * N_SZ + i0 + m) * ID + n] = acc[r] + b_e1[n];
    }
  }

  // ---- v = feats@W_out + pos@W_pos + b_out -> U (reuse) -------------------
  for (int nt = 0; nt < 8; ++nt) {
    v8f acc = {0.f, 0.f, 0.f, 0.f, 0.f, 0.f, 0.f, 0.f};
#pragma unroll
    for (int kk = 0; kk < HD; kk += 4) {
      v2f a; a.x = F[row * HD + kk + kb]; a.y = F[row * HD + kk + kb + 1];
      v2f w; w.x = W_out[(kk + kb    ) * ID + nt * 16 + row];
             w.y = W_out[(kk + kb + 1) * ID + nt * 16 + row];
      acc = wmma_f32(a, w, acc);
    }
#pragma unroll
    for (int r = 0; r < 8; ++r) {
      int m = r + hi * 8, n = nt * 16 + row;
      U[m * ID + n] = acc[r] + b_out[n]
                    + p0[r] * W_pos[n] + p1[r] * W_pos[ID + n];
    }
  }

  // ---- C = U @ W_e1 (no bias) --------------------------------------------
  for (int nt = 0; nt < 8; ++nt) {
    v8f acc = {0.f, 0.f, 0.f, 0.f, 0.f, 0.f, 0.f, 0.f};
#pragma unroll
    for (int kk = 0; kk < ID; kk += 4) {
      v2f a; a.x = U[row * ID + kk + kb]; a.y = U[row * ID + kk + kb + 1];
      v2f w; w.x = W_e1[(kk + kb    ) * ID + nt * 16 + row];
             w.y = W_e1[(kk + kb + 1) * ID + nt * 16 + row];
      acc = wmma_f32(a, w, acc);
    }
#pragma unroll
    for (int r = 0; r < 8; ++r) {
      int m = r + hi * 8, n = nt * 16 + row;
      Cmat[(b * N_SZ + i0 + m) * ID + n] = acc[r];
    }
  }
}

// ---------------------------------------------------------------------------
// pair: node_max[b,j] = max_i mask(i,j) * (sum_k w2[k]*gelu(A[i,k]-C[j,k]) + b_e2)
// Block = 32 j x 128 i tile; 8 lanes per j split the 128 k's (16 each).
// C tile staged memory->LDS with GLOBAL_LOAD_ASYNC_TO_LDS_B128 when available.
// ---------------------------------------------------------------------------
#define JT 32
#define IT 128
__global__ __launch_bounds__(256)
void pair_kernel(const float* __restrict__ Amat, const float* __restrict__ Cmat,
                 const float* __restrict__ emb,  const float* __restrict__ cls,
                 const int* __restrict__ aid,
                 const float* __restrict__ W_e2, const float* __restrict__ b_e2,
                 int* __restrict__ nm_bits)
{
  __shared__ float Cs[JT][ID];
  __shared__ float angj[JT];
  __shared__ float clsj[JT];
  __shared__ int   idj[JT];

  const int b  = blockIdx.z;
  const int j0 = blockIdx.y * JT;
  const int i0 = blockIdx.x * IT;
  const int tid = threadIdx.x;
  const int jl = tid >> 3;      // 0..31
  const int kc = tid & 7;       // 0..7 -> 16 k's each
  const int j  = j0 + jl;

#if HAVE_ASYNC_LDS
  // 32 rows x 128 floats = 1024 float4 chunks; 4 per thread, direct to LDS.
#pragma unroll
  for (int t = 0; t < 4; ++t) {
    int c = tid + t * 256;
    int r = c >> 5, q = c & 31;               // row, float4-column
    const float* g = &Cmat[(b * N_SZ + j0 + r) * ID + q * 4];
    float* l = &Cs[r][q * 4];
    __builtin_amdgcn_global_load_async_to_lds_b128(
        (gptr_v4i)(uintptr_t)g,
        (lptr_v4i)(uint32_t)(uintptr_t)l,
        0, 0);
  }
#else
  for (int idx = tid; idx < JT * ID; idx += 256) {
    Cs[idx >> 7][idx & 127] = Cmat[(b * N_SZ + j0 + (idx >> 7)) * ID + (idx & 127)];
  }
#endif
  if (tid < JT) {
    angj[tid] = emb[(b * N_SZ + j0 + tid) * 2] * PI_F;
    clsj[tid] = cls[b * N_SZ + j0 + tid];
    idj[tid]  = aid[b * N_SZ + j0 + tid];
  }
#if HAVE_ASYNC_LDS
  wait_async0();
#endif
  __syncthreads();

  float w2[16], cl[16];
#pragma unroll
  for (int t = 0; t < 16; ++t) {
    w2[t] = W_e2[kc * 16 + t];      // W_e2 is [128,1]
    cl[t] = Cs[jl][kc * 16 + t];
  }
  const float aj = angj[jl], cj = clsj[jl];
  const int   jid = idj[jl];
  const float be2 = b_e2[0];

  float accm = 0.f;
  for (int i = i0; i < i0 + IT; ++i) {
    const float* Ar = &Amat[(b * N_SZ + i) * ID + kc * 16];
    if (i + 8 < i0 + IT) {
      __builtin_prefetch(&Amat[(b * N_SZ + i + 8) * ID + kc * 16], 0, 1);
    }
    float s = 0.f;
#pragma unroll
    for (int q = 0; q < 4; ++q) {
      float4 av = *(const float4*)(Ar + q * 4);
      s += w2[q*4+0] * gelu_exact(av.x - cl[q*4+0]);
      s += w2[q*4+1] * gelu_exact(av.y - cl[q*4+1]);
      s += w2[q*4+2] * gelu_exact(av.z - cl[q*4+2]);
      s += w2[q*4+3] * gelu_exact(av.w - cl[q*4+3]);
    }
    // butterfly over the 8 k-chunk lanes (contiguous within the wave)
    s += __shfl_xor(s, 1, 32);
    s += __shfl_xor(s, 2, 32);
    s += __shfl_xor(s, 4, 32);
    if (kc == 0) {
      float ai = emb[(b * N_SZ + i) * 2] * PI_F;
      float ci = cls[b * N_SZ + i];
      int   iid = aid[b * N_SZ + i];
      bool sup = (fabsf(ai - aj) < 0.5f) &&
                 ((ci > cj) || ((ci == cj) && (iid > jid)));
      float e = sup ? (s + be2) : 0.f;
      accm = fmaxf(accm, e);
    }
  }
  if (kc == 0) {
    // accm >= 0, so int-bit compare preserves float order
    atomicMax(&nm_bits[b * N_SZ + j], __float_as_int(accm));
  }
}

// ---------------------------------------------------------------------------
// head: scalar -> 64 -> 64 -> 1 MLP + conf threshold + sigmoid
// ---------------------------------------------------------------------------
__global__ __launch_bounds__(256)
void head_kernel(const int* __restrict__ nm_bits, const float* __restrict__ cls,
                 const float* __restrict__ W_n1, const float* __restrict__ b_n1,
                 const float* __restrict__ W_n2, const float* __restrict__ b_n2,
                 const float* __restrict__ W_hd, const float* __restrict__ b_hd,
                 float* __restrict__ out)
{
  int idx = blockIdx.x * blockDim.x + threadIdx.x;
  if (idx >= B_SZ * N_SZ) return;
  float nm = __int_as_float(nm_bits[idx]);

  float n1[HD];
#pragma unroll
  for (int h = 0; h < HD; ++h)
    n1[h] = fmaxf(nm * W_n1[h] + b_n1[h], 0.f);

  float logit = b_hd[0];
  for (int h2 = 0; h2 < HD; ++h2) {
    float s = b_n2[h2];
#pragma unroll 8
    for (int h = 0; h < HD; ++h) s += n1[h] * W_n2[h * HD + h2];
    logit += fmaxf(s, 0.f) * W_hd[h2];
  }
  if (cls[idx] < 0.4f) logit = -1e6f;
  out[idx] = 1.f / (1.f + expf(-logit));
}

extern "C" void kernel_launch(void* const* d_in, const int* in_sizes, int n_in,
                              void* d_out, int out_size, void* d_ws, size_t ws_size,
                              hipStream_t stream) {
  const float* X     = (const float*)d_in[0];
  const float* cls   = (const float*)d_in[1];
  const int*   aid   = (const int*)d_in[2];
  const float* emb   = (const float*)d_in[3];
  const float* W_cls = (const float*)d_in[4];  const float* b_cls = (const float*)d_in[5];
  const float* W_pos = (const float*)d_in[6];  const float* b_pos = (const float*)d_in[7];
  const float* W_in  = (const float*)d_in[8];  const float* b_in  = (const float*)d_in[9];
  const float* W_out = (const float*)d_in[10]; const float* b_out = (const float*)d_in[11];
  const float* W_e1  = (const float*)d_in[12]; const float* b_e1  = (const float*)d_in[13];
  const float* W_e2  = (const float*)d_in[14]; const float* b_e2  = (const float*)d_in[15];
  const float* W_n1  = (const float*)d_in[16]; const float* b_n1  = (const float*)d_in[17];
  const float* W_n2  = (const float*)d_in[18]; const float* b_n2  = (const float*)d_in[19];
  const float* W_hd  = (const float*)d_in[20]; const float* b_hd  = (const float*)d_in[21];
  float* out = (float*)d_out;

  float* Amat = (float*)d_ws;                       // 4*512*128 f32 = 1 MB
  float* Cmat = Amat + B_SZ * N_SZ * ID;            // 1 MB
  int*   nm   = (int*)(Cmat + B_SZ * N_SZ * ID);    // 8 KB

  init_kernel<<<(B_SZ * N_SZ + 255) / 256, 256, 0, stream>>>(nm);
  prep_kernel<<<32, 128, 0, stream>>>(X, emb, W_cls, b_cls, W_pos, b_pos,
                                      W_in, b_in, W_out, b_out, W_e1, b_e1,
                                      Amat, Cmat);
  dim3 pg(N_SZ / IT, N_SZ / JT, B_SZ);
  pair_kernel<<<pg, 256, 0, stream>>>(Amat, Cmat, emb, cls, aid, W_e2, b_e2, nm);
  head_kernel<<<(B_SZ * N_SZ + 255) / 256, 256, 0, stream>>>(nm, cls, W_n1, b_n1,
                                                             W_n2, b_n2, W_hd, b_hd, out);
}